// QLSTMTagger_65481071406493
// MI455X (gfx1250) — compile-verified
//
#include <hip/hip_runtime.h>
#include <hip/hip_bf16.h>
#include <math.h>

#define SEQ   4096
#define EMBED 256
#define NQ    8
#define TAGS  50
#define DIN   264   // EMBED + NQ

typedef float v2f __attribute__((ext_vector_type(2)));
typedef float v8f __attribute__((ext_vector_type(8)));

// ---------------------------------------------------------------------------
// Kernel 1: embedding gather fused with the input-projection GEMM.
//   pre[t][n] = emb[sent[t]] . Wcat[n][0:256] + bias[n],  n = gate*8 + unit
// V_WMMA_F32_16X16X4_F32: M=4096, N=32, K=256 -> 512 wave-tiles, 64 WMMA/tile.
// fp32 end to end (no precision loss vs reference).
// ---------------------------------------------------------------------------
__global__ __launch_bounds__(256)
void qlstm_gate_gemm(const int* __restrict__ sent,
                     const float* __restrict__ emb,
                     const float* __restrict__ Wf, const float* __restrict__ bf,
                     const float* __restrict__ Wi, const float* __restrict__ bi,
                     const float* __restrict__ Wu, const float* __restrict__ bu,
                     const float* __restrict__ Wo, const float* __restrict__ bo,
                     float* __restrict__ pre)
{
  const int lane = threadIdx.x & 31;
  const int wave = threadIdx.x >> 5;
  const int tile = blockIdx.x * (blockDim.x >> 5) + wave;   // 0..511
  const int rowTile = tile >> 1;                            // 16-row tile
  const int colTile = tile & 1;                             // 0:{f,i} 1:{u,o}
  const int half = lane >> 4;                               // K sub-offset half
  const int l15  = lane & 15;

  // A: 16x4 fp32 tile of gathered embedding rows.
  //   lanes 0-15: A[M=l15][k,k+1]; lanes 16-31: A[M=l15][k+2,k+3]
  const int arow_idx = rowTile * 16 + l15;
  const float* arow = emb + (size_t)sent[arow_idx] * EMBED + (half << 1);

  // B: 4x16 fp32 tile of Wcat^T. Column n picks one (gate,unit) weight row.
  const int n = colTile * 16 + l15;            // 0..31
  const int g = n >> 3;
  const int j = n & 7;
  const float* Wg = (g == 0) ? Wf : (g == 1) ? Wi : (g == 2) ? Wu : Wo;
  const float* bg = (g == 0) ? bf : (g == 1) ? bi : (g == 2) ? bu : bo;
  const float* brow = Wg + j * DIN + (half << 1);

  v8f c = {0.f, 0.f, 0.f, 0.f, 0.f, 0.f, 0.f, 0.f};
#pragma unroll 8
  for (int k = 0; k < EMBED; k += 4) {
    v2f a = *(const v2f*)(arow + k);
    v2f b = *(const v2f*)(brow + k);
    c = __builtin_amdgcn_wmma_f32_16x16x4_f32(false, a, false, b,
                                              (short)0, c, false, false);
  }

  // C epilogue: VGPR r holds row (rowBase + r), column l15 (+ bias fold).
  const float bias = bg[j];
  const int rowBase = rowTile * 16 + (half << 3);
#pragma unroll
  for (int r = 0; r < 8; ++r) {
    pre[(size_t)(rowBase + r) * 32 + n] = c[r] + bias;
  }
}

// ---------------------------------------------------------------------------
// Kernel 2: the inherently sequential LSTM scan. One wave32; lane l handles
// gate g=l/8, unit j=l%8. Quantum layer in closed form:
//   z_j = cos^2(angle_j);  out_0 = prod_{1..7} z;  out_k = prod_{0..k} z.
// Latency-optimized: next-step `pre` load software-pipelined one iteration
// ahead; dot product and masked product reduced as depth-3 trees.
// ---------------------------------------------------------------------------
__device__ __forceinline__ float fast_tanh(float x) {
  // 1 - 2/(e^{2x}+1): monotone-safe at +/-inf (no NaN).
  return 1.f - 2.f / (__expf(2.f * x) + 1.f);
}

__global__ __launch_bounds__(32)
void qlstm_scan(const float* __restrict__ pre,
                const float* __restrict__ Wf, const float* __restrict__ Wi,
                const float* __restrict__ Wu, const float* __restrict__ Wo,
                float* __restrict__ hout)
{
  const int l = threadIdx.x;          // 0..31
  const int g = l >> 3, j = l & 7;
  const float* Wg = (g == 0) ? Wf : (g == 1) ? Wi : (g == 2) ? Wu : Wo;

  float wh[NQ];                       // recurrent weights Wcat[n][256:264]
#pragma unroll
  for (int k = 0; k < NQ; ++k) wh[k] = Wg[j * DIN + EMBED + k];

  float hx = 0.f, cx = 0.f;           // state for unit j (replicated per group)
  float cur = pre[l];                 // pipelined pre-activation for step t
  for (int t = 0; t < SEQ; ++t) {
    // Issue next step's load immediately (clamped index = branch-free);
    // its latency hides behind this step's shuffles + transcendentals.
    const int tn = (t + 1 < SEQ) ? (t + 1) : (SEQ - 1);
    const float nxt = pre[tn * 32 + l];

    // angle = pre + Wh . h : 8 independent bpermutes, depth-3 add tree.
    float p[NQ];
#pragma unroll
    for (int k = 0; k < NQ; ++k) p[k] = wh[k] * __shfl(hx, k, 32);
    const float ang = cur + (((p[0] + p[1]) + (p[2] + p[3])) +
                             ((p[4] + p[5]) + (p[6] + p[7])));

    const float cz = __cosf(ang);
    const float z  = cz * cz;

    // masked product over this gate's 8 lanes (branch-free, depth-3 tree)
    float m[NQ];
#pragma unroll
    for (int k = 0; k < NQ; ++k) {
      const float zk = __shfl(z, (g << 3) + k, 32);
      const bool inc = (j == 0) ? (k >= 1) : (k <= j);
      m[k] = inc ? zk : 1.f;
    }
    const float q = ((m[0] * m[1]) * (m[2] * m[3])) *
                    ((m[4] * m[5]) * (m[6] * m[7]));

    // gate nonlinearity: sigmoid for f,i,o; tanh for u (g==2)
    const float sig = 1.f / (1.f + __expf(-q));
    const float act = (g == 2) ? fast_tanh(q) : sig;

    // combine: every lane rebuilds unit-j state redundantly
    const float fj = __shfl(act, j,      32);
    const float ij = __shfl(act, 8 + j,  32);
    const float gj = __shfl(act, 16 + j, 32);
    const float oj = __shfl(act, 24 + j, 32);
    cx = fmaf(fj, cx, ij * gj);
    hx = oj * fast_tanh(cx);

    if (l < NQ) hout[t * NQ + l] = hx;   // fire-and-forget store
    cur = nxt;
  }
}

// ---------------------------------------------------------------------------
// Kernel 3: tag projection + log-softmax. One thread per timestep.
// ---------------------------------------------------------------------------
__global__ __launch_bounds__(256)
void qlstm_tagger(const float* __restrict__ hseq,
                  const float* __restrict__ Wt, const float* __restrict__ bt,
                  float* __restrict__ out)
{
  const int t = blockIdx.x * blockDim.x + threadIdx.x;
  if (t >= SEQ) return;
  float h[NQ];
#pragma unroll
  for (int k = 0; k < NQ; ++k) h[k] = hseq[t * NQ + k];

  float logits[TAGS];
  float mx = -3.402823466e38f;
  for (int c = 0; c < TAGS; ++c) {
    float a = bt[c];
#pragma unroll
    for (int k = 0; k < NQ; ++k) a = fmaf(h[k], Wt[c * NQ + k], a);
    logits[c] = a;
    mx = fmaxf(mx, a);
  }
  float s = 0.f;
  for (int c = 0; c < TAGS; ++c) s += __expf(logits[c] - mx);
  const float lse = mx + __logf(s);
  for (int c = 0; c < TAGS; ++c)
    out[(size_t)t * TAGS + c] = logits[c] - lse;
}

// ---------------------------------------------------------------------------
extern "C" void kernel_launch(void* const* d_in, const int* in_sizes, int n_in,
                              void* d_out, int out_size, void* d_ws, size_t ws_size,
                              hipStream_t stream)
{
  const int*   sent = (const int*)  d_in[0];
  const float* emb  = (const float*)d_in[1];
  const float* Wf   = (const float*)d_in[2];
  const float* bf   = (const float*)d_in[3];
  const float* Wi   = (const float*)d_in[4];
  const float* bi   = (const float*)d_in[5];
  const float* Wu   = (const float*)d_in[6];
  const float* bu   = (const float*)d_in[7];
  const float* Wo   = (const float*)d_in[8];
  const float* bo   = (const float*)d_in[9];
  const float* Wt   = (const float*)d_in[10];
  const float* bt   = (const float*)d_in[11];
  float* out = (float*)d_out;

  float* pre  = (float*)d_ws;                // SEQ*32 f32 (512 KB)
  float* hout = pre + (size_t)SEQ * 32;      // SEQ*NQ f32 (128 KB)

  qlstm_gate_gemm<<<64, 256, 0, stream>>>(sent, emb, Wf, bf, Wi, bi,
                                          Wu, bu, Wo, bo, pre);
  qlstm_scan<<<1, 32, 0, stream>>>(pre, Wf, Wi, Wu, Wo, hout);
  qlstm_tagger<<<SEQ / 256, 256, 0, stream>>>(hout, Wt, bt, out);
}